// CombSubFast_63118839382097
// MI455X (gfx1250) — compile-verified
//
#include <hip/hip_runtime.h>

#define SR_F     44100.0f
#define BLK      512
#define NB       8
#define NF       2048
#define NSAMP    (NF * BLK)      /* 1048576 */
#define NFRAME   (NF + 1)        /* 2049    */
#define NBIN     513
#define PI_F     3.14159265358979f
#define TWO_PI_F 6.28318530717959f

typedef __attribute__((ext_vector_type(2))) float v2f;
typedef __attribute__((ext_vector_type(8))) float v8f;

// ---------------------------------------------------------------------------
// CDNA5 async global->LDS DMA (bypasses VGPR file, tracked by ASYNCcnt).
// SADDR/GVS form: mem = SADDR(64b, uniform) + VADDR(32b, per-lane) + IOFFSET.
// VDST VGPR holds the per-lane LDS byte address.
// ---------------------------------------------------------------------------
__device__ __forceinline__ void async_load_b128(uint32_t lds_addr,
                                                uint64_t gbase,
                                                uint32_t voff) {
  asm volatile("global_load_async_to_lds_b128 %0, %1, %2"
               :: "v"(lds_addr), "v"(voff), "s"(gbase)
               : "memory");
}
__device__ __forceinline__ void wait_asynccnt0() {
  asm volatile("s_wait_asynccnt 0" ::: "memory");
}

// ---------------------------------------------------------------------------
// Kernel 1: per-row inclusive scan of f0/SR -> wrapped rotation -> combtooth
// and per-frame phase output. One block per batch row, chunked Hillis-Steele
// scan with a wrapped carry (rot - round(rot) is invariant to integer shifts,
// so wrapping the carry each chunk preserves f32 precision over 1M samples).
// ---------------------------------------------------------------------------
__global__ void __launch_bounds__(512)
phase_scan_kernel(const float* __restrict__ f0_frames,
                  float* __restrict__ combtooth,
                  float* __restrict__ phase_out) {
  const int b   = blockIdx.x;
  const int tid = threadIdx.x;
  __shared__ float s_val[512];
  __shared__ float s_carry;
  if (tid == 0) s_carry = 0.0f;
  __syncthreads();

  const float* f0row = f0_frames + (size_t)b * NF;
  float*       crow  = combtooth + (size_t)b * NSAMP;

  for (int chunk = 0; chunk < NSAMP / 512; ++chunk) {
    const int   i    = (chunk << 9) + tid;
    const int   k    = i >> 9;
    const float frac = (float)(i & (BLK - 1)) * (1.0f / (float)BLK);
    const float fa   = f0row[k];
    const float fb   = (k + 1 < NF) ? f0row[k + 1] : f0row[NF - 1];
    const float f0   = fa * (1.0f - frac) + fb * frac;

    s_val[tid] = f0 * (1.0f / SR_F);
    __syncthreads();
    for (int off = 1; off < 512; off <<= 1) {
      const float t = (tid >= off) ? s_val[tid - off] : 0.0f;
      __syncthreads();
      s_val[tid] += t;
      __syncthreads();
    }
    const float rot = s_carry + s_val[tid];
    const float rw  = rot - rintf(rot);

    const float x  = SR_F * rw / (f0 + 0.001f);
    float sc = (x == 0.0f) ? 1.0f : __sinf(PI_F * x) / (PI_F * x);
    if (f0 <= 0.0f) sc = 0.0f;
    crow[i] = sc;
    if ((i & (BLK - 1)) == 0)
      phase_out[(size_t)b * NF + (i >> 9)] = TWO_PI_F * rw;

    __syncthreads();
    if (tid == 0) {
      const float nc = s_carry + s_val[511];
      s_carry = nc - rintf(nc);           // wrap carry mod 1
    }
    __syncthreads();
  }
}

// ---------------------------------------------------------------------------
// WMMA helpers: D(16x16) += sign * P_tile(16x4) x Q_tile(4x16), f32.
// A layout (16x4): lanes 0-15 rows M=0..15 hold K={0,1}; lanes 16-31 K={2,3}.
// B layout (4x16): VGPR0 rows K={0|2}, VGPR1 rows K={1|3} (low/high lanes).
// C/D layout: VGPR r -> (M=r, lanes 0-15) / (M=r+8, lanes 16-31).
// ---------------------------------------------------------------------------
__device__ __forceinline__ v8f tile_mm(const float* __restrict__ P,
                                       const float* __restrict__ Q,
                                       float psign, v8f acc,
                                       int lm, int lh, int ti, int tj) {
#pragma unroll
  for (int k0 = 0; k0 < 32; k0 += 4) {
    const int ak = k0 + lh * 2;
    v2f a, bb;
    a.x  = psign * P[(ti * 16 + lm) * 32 + ak];
    a.y  = psign * P[(ti * 16 + lm) * 32 + ak + 1];
    bb.x = Q[ak * 32 + tj * 16 + lm];
    bb.y = Q[(ak + 1) * 32 + tj * 16 + lm];
    acc = __builtin_amdgcn_wmma_f32_16x16x4_f32(false, a, false, bb,
                                                (short)0, acc, false, false);
  }
  return acc;
}

__device__ __forceinline__ void tile_store(float* __restrict__ D, v8f acc,
                                           int lm, int lh, int ti, int tj) {
#pragma unroll
  for (int r = 0; r < 8; ++r)
    D[(ti * 16 + lh * 8 + r) * 32 + tj * 16 + lm] = acc[r];
}

// ---------------------------------------------------------------------------
// Four-step 1024-pt complex FFT: Z viewed as 32x32 (n = n1 + 32*n2, stored
// Z[n1][n2]).  Step1: G = Z x W32 (row DFTs).  Step2: twiddle W1024^(n1*k2)
// applied in registers.  Step3: X = W32 x G, result lands at flat k = 32*k1+k2
// i.e. natural order.  W32 = C - iS with C,S the cos/sin tables in LDS.
// ---------------------------------------------------------------------------
__device__ __forceinline__ void fft1024_ws(float* Zr, float* Zi,
                                           float* Gr, float* Gi,
                                           const float* Cm, const float* Sm,
                                           int lm, int lh, int ti, int tj) {
  // Step 1: Gr = Zr*C + Zi*S ; Gi = Zi*C - Zr*S
  v8f ar = {}, ai = {};
  ar = tile_mm(Zr, Cm,  1.0f, ar, lm, lh, ti, tj);
  ar = tile_mm(Zi, Sm,  1.0f, ar, lm, lh, ti, tj);
  ai = tile_mm(Zi, Cm,  1.0f, ai, lm, lh, ti, tj);
  ai = tile_mm(Zr, Sm, -1.0f, ai, lm, lh, ti, tj);
  // Step 2: twiddle exp(-2*pi*i*row*col/1024) in registers
#pragma unroll
  for (int r = 0; r < 8; ++r) {
    const int row = ti * 16 + lh * 8 + r;
    const int col = tj * 16 + lm;
    float sn, cs;
    __sincosf((TWO_PI_F / 1024.0f) * (float)(row * col), &sn, &cs);
    const float gr = ar[r], gi = ai[r];
    ar[r] = gr * cs + gi * sn;
    ai[r] = gi * cs - gr * sn;
  }
  tile_store(Gr, ar, lm, lh, ti, tj);
  tile_store(Gi, ai, lm, lh, ti, tj);
  __syncthreads();
  // Step 3: Zr = C*Gr + S*Gi ; Zi = C*Gi - S*Gr
  v8f br = {}, bi = {};
  br = tile_mm(Cm, Gr,  1.0f, br, lm, lh, ti, tj);
  br = tile_mm(Sm, Gi,  1.0f, br, lm, lh, ti, tj);
  bi = tile_mm(Cm, Gi,  1.0f, bi, lm, lh, ti, tj);
  bi = tile_mm(Sm, Gr, -1.0f, bi, lm, lh, ti, tj);
  tile_store(Zr, br, lm, lh, ti, tj);
  tile_store(Zi, bi, lm, lh, ti, tj);
  __syncthreads();
}

// ---------------------------------------------------------------------------
// Kernel 2: one block per (batch, frame).  Async-DMA the raw comb+noise frame
// into an LDS staging buffer (overlapped with DFT32 table generation), pack
// as one complex frame, FFT, Hermitian-split into the two real spectra, apply
// filters, Hermitian-extend, inverse FFT via conj trick, window, overlap-add.
// ---------------------------------------------------------------------------
__global__ void __launch_bounds__(128)
synth_kernel(const float* __restrict__ comb,
             const float* __restrict__ noise_u,
             const float* __restrict__ harmo_mag,
             const float* __restrict__ harmo_phase,
             const float* __restrict__ noise_mag,
             float* __restrict__ out_sig) {
  __shared__ float Zr[1024], Zi[1024], Gr[1024], Gi[1024], Cm[1024], Sm[1024];
  __shared__ float Stage[2048];   // [0:1024) raw comb, [1024:2048) raw noise_u

  const int tid  = threadIdx.x;
  const int lane = tid & 31;
  const int wv   = tid >> 5;
  const int lm = lane & 15, lh = lane >> 4;
  const int ti = wv >> 1,   tj = wv & 1;

  const int bidx = blockIdx.x / NFRAME;
  const int j    = blockIdx.x % NFRAME;
  const int jf   = (j < NF) ? j : (NF - 1);
  const size_t fbase = ((size_t)bidx * NF + jf) * NBIN;
  const size_t srow  = (size_t)bidx * NSAMP;

  __builtin_prefetch(&harmo_mag[fbase], 0, 1);
  __builtin_prefetch(&harmo_phase[fbase], 0, 1);
  __builtin_prefetch(&noise_mag[fbase], 0, 1);

  // ---- issue async DMA of the frame into LDS staging ----------------------
  // 16B groups; hop (512) and frame length (1024) are 4-aligned, so the
  // zero-padding boundary never splits a group.  Out-of-range groups are
  // zero-filled via ds_store (disjoint LDS addresses -> no DScnt/ASYNCcnt
  // ordering hazard; both are settled by the wait + barrier below).
  {
    const uint32_t stage_base = (uint32_t)(uintptr_t)(&Stage[0]);
    const uint64_t cbase = (uint64_t)(uintptr_t)(comb + srow);
    const uint64_t nbase = (uint64_t)(uintptr_t)(noise_u + srow);
#pragma unroll
    for (int q = 0; q < 2; ++q) {
      const int e = (tid + 128 * q) * 4;           // element index 0..1020
      const int g = j * BLK + e - BLK;             // global sample index
      if (g >= 0 && g + 3 < NSAMP) {
        async_load_b128(stage_base + (uint32_t)(e * 4),        cbase,
                        (uint32_t)(g * 4));
        async_load_b128(stage_base + 4096u + (uint32_t)(e * 4), nbase,
                        (uint32_t)(g * 4));
      } else {
#pragma unroll
        for (int u = 0; u < 4; ++u) {
          Stage[e + u]        = 0.0f;
          Stage[1024 + e + u] = 0.0f;
        }
      }
    }
  }

  // ---- overlap: generate DFT32 tables while the DMA is in flight ----------
  for (int m = tid; m < 1024; m += 128) {
    const int r = m >> 5, c = m & 31;
    float sn, cs;
    __sincosf((TWO_PI_F / 32.0f) * (float)(r * c), &sn, &cs);
    Cm[m] = cs; Sm[m] = sn;
  }

  wait_asynccnt0();      // this wave's DMAs have landed in LDS
  __syncthreads();       // all waves' DMAs + zero-fills visible

  // Window (sqrt-Hann = sin(pi*t/N)) + transpose into four-step layout:
  // slot = (t%32)*32 + t/32.  Noise padding is applied AFTER 2u-1 in the
  // reference, so re-check the range predicate here.
  for (int t = tid; t < 1024; t += 128) {
    const int  g   = j * BLK + t - BLK;
    const bool inr = (g >= 0 && g < NSAMP);
    const float ct = Stage[t];                                  // 0 if padded
    const float nz = inr ? (Stage[1024 + t] * 2.0f - 1.0f) : 0.0f;
    const float w  = __sinf(PI_F * (float)t * (1.0f / 1024.0f));
    const int   p  = (t & 31) * 32 + (t >> 5);
    Zr[p] = ct * w;
    Zi[p] = nz * w;
  }
  __syncthreads();

  fft1024_ws(Zr, Zi, Gr, Gi, Cm, Sm, lm, lh, ti, tj);   // spectrum in Z

  // Hermitian split of packed FFT, filter, Hermitian-extend into G
  for (int k = tid; k < NBIN; k += 128) {
    const float a  = Zr[k],  bI = Zi[k];
    const int   kn = (1024 - k) & 1023;
    const float c  = Zr[kn], d  = Zi[kn];
    const float ctr =  0.5f * (a + c);
    const float cti =  0.5f * (bI - d);
    const float nzr =  0.5f * (bI + d);
    const float nzi = -0.5f * (a - c);
    const float em = __expf(harmo_mag[fbase + k]);
    float sp, cp;
    __sincosf(PI_F * harmo_phase[fbase + k], &sp, &cp);
    const float hr = em * cp, hi = em * sp;
    const float nfv = __expf(noise_mag[fbase + k]) * (1.0f / 128.0f);
    const float yr = ctr * hr - cti * hi + nzr * nfv;
    const float yi = ctr * hi + cti * hr + nzi * nfv;
    Gr[k] = yr; Gi[k] = yi;
    if (k >= 1 && k <= 511) { Gr[1024 - k] = yr; Gi[1024 - k] = -yi; }
  }
  __syncthreads();

  // Inverse via conjugation: irfft = Re(fft(conj(Y)))/N.  Transpose-load.
  for (int m = tid; m < 1024; m += 128) {
    const int p = (m & 31) * 32 + (m >> 5);
    Zr[p] =  Gr[m];
    Zi[p] = -Gi[m];
  }
  __syncthreads();

  fft1024_ws(Zr, Zi, Gr, Gi, Cm, Sm, lm, lh, ti, tj);   // time in Zr (xN)

  // Window + overlap-add (each output sample touched by exactly 2 frames)
  for (int t = tid; t < 1024; t += 128) {
    const float w   = __sinf(PI_F * (float)t * (1.0f / 1024.0f));
    const float val = Zr[t] * (1.0f / 1024.0f) * w;
    const int   s   = j * BLK + t - BLK;
    if (s >= 0 && s < NSAMP) atomicAdd(&out_sig[srow + s], val);
  }
}

// ---------------------------------------------------------------------------
extern "C" void kernel_launch(void* const* d_in, const int* in_sizes, int n_in,
                              void* d_out, int out_size, void* d_ws, size_t ws_size,
                              hipStream_t stream) {
  (void)in_sizes; (void)n_in; (void)out_size; (void)ws_size;
  const float* f0 = (const float*)d_in[0];
  const float* hm = (const float*)d_in[1];
  const float* hp = (const float*)d_in[2];
  const float* nm = (const float*)d_in[3];
  const float* nu = (const float*)d_in[4];

  float* sig   = (float*)d_out;
  float* phase = sig + (size_t)NB * NSAMP;   // phase_frames output (B*F)
  float* combw = (float*)d_ws;               // combtooth scratch (B*NSAMP f32)

  hipMemsetAsync(sig, 0, (size_t)NB * NSAMP * sizeof(float), stream);
  phase_scan_kernel<<<NB, 512, 0, stream>>>(f0, combw, phase);
  synth_kernel<<<NB * NFRAME, 128, 0, stream>>>(combw, nu, hm, hp, nm, sig);
}